// SelfStructure_31293131718891
// MI455X (gfx1250) — compile-verified
//
#include <hip/hip_runtime.h>

typedef __attribute__((ext_vector_type(2))) float v2f;
typedef __attribute__((ext_vector_type(4))) float v4f;
typedef __attribute__((ext_vector_type(8))) float v8f;

#define B_   8
#define C_   1024
#define HW   48
#define N_   2304     // 48*48
#define K_   32
#define CG   32
#define RM   384
#define NEG_INF (-3.402823466e38f)

// ---------------- kernel 1: bilinear 384->48 (half-pixel, scale 8 => avg of 2x2 at 8d+3) + threshold
__global__ __launch_bounds__(256) void k_mask(const float* __restrict__ ref,
                                              float* __restrict__ mask) {
    int t = blockIdx.x * 256 + threadIdx.x;           // B_*N_ threads
    int b = t / N_;
    int m = t - b * N_;
    int y = m / HW, x = m - y * HW;
    int y0 = 8 * y + 3, x0 = 8 * x + 3;
    const float* p = ref + (size_t)b * (RM * RM);
    float v = 0.25f * (p[y0 * RM + x0] + p[y0 * RM + x0 + 1] +
                       p[(y0 + 1) * RM + x0] + p[(y0 + 1) * RM + x0 + 1]);
    mask[t] = (v > 0.5f) ? 1.0f : 0.0f;
}

// ---------------- kernel 2: scores[b,n] = corr[b,n,:] . mask[b,:]
// 170 MB stream-once => non-temporal loads on corr; mask stays cache-resident.
__global__ __launch_bounds__(256) void k_scores(const float* __restrict__ corr,
                                                const float* __restrict__ mask,
                                                float* __restrict__ scores) {
    int wave = (blockIdx.x * 256 + threadIdx.x) >> 5;  // one wave per row, B_*N_ rows
    int lane = threadIdx.x & 31;
    int b = wave / N_;
    const v4f* cr = (const v4f*)(corr + (size_t)wave * N_);
    const v4f* mk = (const v4f*)(mask + (size_t)b * N_);
    float acc = 0.f;
#pragma unroll 2
    for (int j = lane; j < N_ / 4; j += 32) {
        v4f c4 = __builtin_nontemporal_load(cr + j);
        v4f m4 = mk[j];
        acc += c4.x * m4.x + c4.y * m4.y + c4.z * m4.z + c4.w * m4.w;
    }
#pragma unroll
    for (int off = 16; off; off >>= 1) acc += __shfl_xor(acc, off, 32);
    if (lane == 0) scores[wave] = acc;
}

// ---------------- kernel 3: top-32 (descending, ties -> lower index) per batch, one block per batch
__global__ __launch_bounds__(256) void k_topk(const float* __restrict__ scores,
                                              int* __restrict__ idx) {
    __shared__ float sv[N_];
    __shared__ float rv[256];
    __shared__ int   ri[256];
    int b = blockIdx.x, tid = threadIdx.x;
    for (int i = tid; i < N_; i += 256) sv[i] = scores[(size_t)b * N_ + i];
    __syncthreads();
    for (int it = 0; it < K_; it++) {
        float bv = NEG_INF; int bi = 0x7fffffff;
        for (int i = tid; i < N_; i += 256) {
            float v = sv[i];
            if (v > bv || (v == bv && i < bi)) { bv = v; bi = i; }
        }
        rv[tid] = bv; ri[tid] = bi;
        __syncthreads();
        for (int off = 128; off > 0; off >>= 1) {
            if (tid < off) {
                float v2 = rv[tid + off]; int i2 = ri[tid + off];
                if (v2 > rv[tid] || (v2 == rv[tid] && i2 < ri[tid])) { rv[tid] = v2; ri[tid] = i2; }
            }
            __syncthreads();
        }
        if (tid == 0) { idx[b * K_ + it] = ri[0]; sv[ri[0]] = NEG_INF; }
        __syncthreads();
    }
}

// ---------------- kernel 4: gather struct[b,k,c] = cur[b,c,idx[b,k]]
__global__ __launch_bounds__(256) void k_gather(const float* __restrict__ cur,
                                                const int* __restrict__ idx,
                                                float* __restrict__ sbuf) {
    int t = blockIdx.x * 256 + threadIdx.x;   // B_*K_*C_ threads, c fastest
    int c = t & (C_ - 1);
    int k = (t >> 10) & (K_ - 1);
    int b = t >> 15;
    int n = idx[b * K_ + k];
    sbuf[t] = cur[((size_t)b * C_ + c) * N_ + n];
}

// ---------------- kernel 5: struct_info[b] = S[32x1024] x X[1024x2304], wave = 32x16 tile (v_wmma_f32_16x16x4_f32)
__global__ __launch_bounds__(256) void k_structinfo(const float* __restrict__ S,
                                                    const float* __restrict__ X,
                                                    float* __restrict__ SI) {
    int wave = (blockIdx.x * 256 + threadIdx.x) >> 5;  // 1152 waves exactly
    int lane = threadIdx.x & 31;
    int b  = wave / (N_ / 16);
    int nt = wave - b * (N_ / 16);
    int n0 = nt * 16;
    bool hi = lane >= 16;
    int l16 = lane & 15;
    int ih = hi ? 2 : 0;
    int col = n0 + l16;
    const float* Sb = S + (size_t)b * (K_ * C_);
    const float* Xb = X + (size_t)b * (C_ * N_);
    v8f c0 = {}, c1 = {};
#pragma unroll 2
    for (int c = 0; c < C_; c += 4) {
        v2f a0 = *(const v2f*)(Sb + (size_t)l16 * C_ + c + ih);
        v2f a1 = *(const v2f*)(Sb + (size_t)(16 + l16) * C_ + c + ih);
        v2f bt;
        bt.x = Xb[(size_t)(c + ih) * N_ + col];
        bt.y = Xb[(size_t)(c + ih + 1) * N_ + col];
        c0 = __builtin_amdgcn_wmma_f32_16x16x4_f32(false, a0, false, bt, (short)0, c0, false, false);
        c1 = __builtin_amdgcn_wmma_f32_16x16x4_f32(false, a1, false, bt, (short)0, c1, false, false);
    }
    float* out = SI + (size_t)b * (K_ * N_);
    int mb = hi ? 8 : 0;
#pragma unroll
    for (int r = 0; r < 8; r++) {
        out[(size_t)(mb + r) * N_ + col]      = c0[r];
        out[(size_t)(16 + mb + r) * N_ + col] = c1[r];
    }
}

// ---------------- kernel 6 (fused): grouped conv path AND global conv path share every X-tile load.
//   group:  out[b, o,    n] = sum_k relu(W_k X_k + b_k)[o,n] * SI[b,k,n]
//   global: out[b, 32+o, n] = relu(Wg X + bg)[o,n] * mean_k SI[b,k,n]
// Per wave: one (b, 16-col tile); k-loop over 32 groups; per i-step 4 WMMA sharing one B operand.
__global__ __launch_bounds__(256) void k_fused(const float* __restrict__ cur,
                                               const float* __restrict__ gw,
                                               const float* __restrict__ gb,
                                               const float* __restrict__ glw,
                                               const float* __restrict__ glb,
                                               const float* __restrict__ SI,
                                               float* __restrict__ out) {
    int wave = (blockIdx.x * 256 + threadIdx.x) >> 5;
    int lane = threadIdx.x & 31;
    int b  = wave / (N_ / 16);
    int nt = wave - b * (N_ / 16);
    int n0 = nt * 16;
    bool hi = lane >= 16;
    int l16 = lane & 15;
    int ih = hi ? 2 : 0;
    int col = n0 + l16;
    int ob = hi ? 8 : 0;
    const float* Xb  = cur + (size_t)b * (C_ * N_);
    const float* SIb = SI  + (size_t)b * (K_ * N_);
    float acc0[8] = {0.f, 0.f, 0.f, 0.f, 0.f, 0.f, 0.f, 0.f};
    float acc1[8] = {0.f, 0.f, 0.f, 0.f, 0.f, 0.f, 0.f, 0.f};
    v8f g0 = {}, g1 = {};          // global-conv accumulators (full 1024-deep K loop)
    float ssum = 0.f;
    for (int k = 0; k < K_; k++) {
        const float* Wk = gw + (size_t)k * (CG * CG);
        const float* Xk = Xb + (size_t)(k * CG) * N_;
        v8f c0 = {}, c1 = {};
#pragma unroll
        for (int i = 0; i < CG; i += 4) {
            v2f a0  = *(const v2f*)(Wk + (size_t)l16 * CG + i + ih);
            v2f a1  = *(const v2f*)(Wk + (size_t)(16 + l16) * CG + i + ih);
            v2f ga0 = *(const v2f*)(glw + (size_t)l16 * C_ + k * CG + i + ih);
            v2f ga1 = *(const v2f*)(glw + (size_t)(16 + l16) * C_ + k * CG + i + ih);
            v2f bt;
            bt.x = Xk[(size_t)(i + ih) * N_ + col];
            bt.y = Xk[(size_t)(i + ih + 1) * N_ + col];
            c0 = __builtin_amdgcn_wmma_f32_16x16x4_f32(false, a0,  false, bt, (short)0, c0, false, false);
            c1 = __builtin_amdgcn_wmma_f32_16x16x4_f32(false, a1,  false, bt, (short)0, c1, false, false);
            g0 = __builtin_amdgcn_wmma_f32_16x16x4_f32(false, ga0, false, bt, (short)0, g0, false, false);
            g1 = __builtin_amdgcn_wmma_f32_16x16x4_f32(false, ga1, false, bt, (short)0, g1, false, false);
        }
        float s = SIb[(size_t)k * N_ + col];
        ssum += s;
#pragma unroll
        for (int r = 0; r < 8; r++) {
            float v0 = c0[r] + gb[k * CG + ob + r];
            float v1 = c1[r] + gb[k * CG + 16 + ob + r];
            acc0[r] += fmaxf(v0, 0.f) * s;
            acc1[r] += fmaxf(v1, 0.f) * s;
        }
    }
    float mean = ssum * (1.0f / 32.0f);
#pragma unroll
    for (int r = 0; r < 8; r++) {
        int o0 = ob + r, o1 = 16 + ob + r;
        out[((size_t)b * 64 + o0) * N_ + col] = acc0[r];
        out[((size_t)b * 64 + o1) * N_ + col] = acc1[r];
        out[((size_t)b * 64 + 32 + o0) * N_ + col] = fmaxf(g0[r] + glb[o0], 0.f) * mean;
        out[((size_t)b * 64 + 32 + o1) * N_ + col] = fmaxf(g1[r] + glb[o1], 0.f) * mean;
    }
}

extern "C" void kernel_launch(void* const* d_in, const int* in_sizes, int n_in,
                              void* d_out, int out_size, void* d_ws, size_t ws_size,
                              hipStream_t stream) {
    const float* corr = (const float*)d_in[0];   // [8,48,48,48,48]
    const float* cur  = (const float*)d_in[1];   // [8,1024,48,48]
    const float* refm = (const float*)d_in[2];   // [8,1,384,384]
    const float* gw   = (const float*)d_in[3];   // [32,32,32]
    const float* gb   = (const float*)d_in[4];   // [1024]
    const float* glw  = (const float*)d_in[5];   // [32,1024]
    const float* glb  = (const float*)d_in[6];   // [32]
    float* out = (float*)d_out;                  // [8,64,48,48]

    // workspace layout (all 256B-aligned): mask | scores | idx | struct | struct_info  (~3.6 MB)
    char* ws = (char*)d_ws;
    float* mask   = (float*)(ws);                       // 8*2304 f
    float* scores = (float*)(ws + 73728);               // 8*2304 f
    int*   idx    = (int*)  (ws + 147456);              // 8*32 i
    float* sbuf   = (float*)(ws + 148480);              // 8*32*1024 f
    float* sinfo  = (float*)(ws + 148480 + 1048576);    // 8*32*2304 f

    k_mask      <<<(B_ * N_) / 256, 256, 0, stream>>>(refm, mask);
    k_scores    <<<(B_ * N_) / 8,   256, 0, stream>>>(corr, mask, scores);
    k_topk      <<<B_,              256, 0, stream>>>(scores, idx);
    k_gather    <<<(B_ * K_ * C_) / 256, 256, 0, stream>>>(cur, idx, sbuf);
    k_structinfo<<<(B_ * (N_ / 16)) / 8, 256, 0, stream>>>(sbuf, cur, sinfo);
    k_fused     <<<(B_ * (N_ / 16)) / 8, 256, 0, stream>>>(cur, gw, gb, glw, glb, sinfo, out);
}